// GraphTransformerAttention_5325759447050
// MI455X (gfx1250) — compile-verified
//
#include <hip/hip_runtime.h>
#include <cstddef>

typedef float v2f __attribute__((ext_vector_type(2)));
typedef float v8f __attribute__((ext_vector_type(8)));

constexpr int NHEADS = 8;
constexpr int ODIM   = 16;
constexpr int HD     = NHEADS * ODIM;  // 128
constexpr int INDIM  = 128;

// ---------------------------------------------------------------------------
// Zero-init z [N*H] and out [N*H*D] (out is the wV accumulator).
// ---------------------------------------------------------------------------
__global__ void zero_kernel(float* __restrict__ z, float* __restrict__ out,
                            int nz, int no) {
    int i = blockIdx.x * blockDim.x + threadIdx.x;
    if (i < nz) z[i] = 0.0f;
    if (i < no) out[i] = 0.0f;
}

// ---------------------------------------------------------------------------
// One-off transpose of the three 128x128 weight matrices:
//   Wt[m][c][k] = W_m[k][c]
// so the WMMA B-operand pair (W[ka][col], W[ka+1][col]) becomes one
// contiguous 8-byte load (global_load_b64) in the GEMM inner loop.
// ---------------------------------------------------------------------------
__global__ void transpose_w_kernel(const float* __restrict__ Wq,
                                   const float* __restrict__ Wk,
                                   const float* __restrict__ Wv,
                                   float* __restrict__ Wt) {
    int i = blockIdx.x * blockDim.x + threadIdx.x;   // over 3*INDIM*HD
    if (i >= 3 * INDIM * HD) return;
    int m = i / (INDIM * HD);
    int r = i % (INDIM * HD);
    int c = r / INDIM;          // column of original W
    int k = r % INDIM;          // row of original W
    const float* W = (m == 0) ? Wq : (m == 1) ? Wk : Wv;
    Wt[i] = W[k * HD + c];
}

// ---------------------------------------------------------------------------
// QKV projection: out[n,c] = sum_k h[n,k] * W[k,c] + b[c]
// blockIdx.y selects (Wt_q,bq,Q) / (Wt_k,bk,K) / (Wt_v,bv,V).
// One wave32 computes a 16-row strip across all 8 column tiles using
// V_WMMA_F32_16X16X4_F32 (f32 A/B, f32 accumulate — matches the reference).
//
// VGPR layouts per CDNA5 ISA 7.12.2 (f32):
//   A 16x4 : lane<16 -> M=lane,    VGPR0/1 = K {k, k+1}
//            lane>=16-> M=lane-16, VGPR0/1 = K {k+2, k+3}
//   B 4x16 : lane<16 -> N=lane,    VGPR0/1 = K rows {k, k+1}
//            lane>=16-> N=lane-16, VGPR0/1 = K rows {k+2, k+3}
//   C/D    : VGPR i, lane l -> (M = i + 8*(l>=16), N = l%16)
// ---------------------------------------------------------------------------
__global__ __launch_bounds__(256) void qkv_gemm_kernel(
    const float* __restrict__ h,
    const float* __restrict__ Wt,      // [3][HD][INDIM] transposed weights
    const float* __restrict__ bq, const float* __restrict__ bk,
    const float* __restrict__ bv,
    float* __restrict__ Qo, float* __restrict__ Ko, float* __restrict__ Vo,
    int N)
{
    const float* bias; float* out;
    if (blockIdx.y == 0)      { bias = bq; out = Qo; }
    else if (blockIdx.y == 1) { bias = bk; out = Ko; }
    else                      { bias = bv; out = Vo; }
    const float* __restrict__ Wsel = Wt + (size_t)blockIdx.y * INDIM * HD;

    const int lane = threadIdx.x & 31;
    const int wave = threadIdx.x >> 5;
    const int tile = blockIdx.x * 8 + wave;           // 16-row strip index
    const int tiles = (N + 15) >> 4;
    if (tile >= tiles) return;                        // wave-uniform: EXEC stays full

    const int row0 = tile * 16;
    const int lr   = lane & 15;                       // M (for A) / N (for B,C)
    const int hb   = lane >> 4;                       // lane-half selects K sub-block

    // Clamp A-row reads for a possible ragged tail (stores are guarded below),
    // so every WMMA still executes with EXEC == all ones.
    const int arow = min(row0 + lr, N - 1);
    const float* __restrict__ hrow = h + (size_t)arow * INDIM;

    // Per-lane contiguous B stream: lane reads Wt row `col` (512 B / k-loop).
    const float* __restrict__ wt0 = Wsel + (size_t)lr * INDIM;   // col tile 0

    v8f acc[8];
    #pragma unroll
    for (int t = 0; t < 8; ++t) acc[t] = (v8f)0.0f;

    for (int k = 0; k < INDIM; k += 4) {
        const int ka = k + 2 * hb;                    // 8B-aligned (ka even)
        const v2f a = *(const v2f*)(hrow + ka);
        #pragma unroll
        for (int t = 0; t < 8; ++t) {
            // Wt[col][ka], Wt[col][ka+1] are contiguous -> one b64 load.
            const v2f wb = *(const v2f*)(wt0 + (size_t)t * 16 * INDIM + ka);
            acc[t] = __builtin_amdgcn_wmma_f32_16x16x4_f32(
                false, a, false, wb, (short)0, acc[t], false, false);
        }
    }

    // Epilogue: add bias (per output column), store D tiles.
    #pragma unroll
    for (int t = 0; t < 8; ++t) {
        const int col = t * 16 + lr;
        const float bb = bias[col];
        #pragma unroll
        for (int i = 0; i < 8; ++i) {
            const int row = row0 + i + 8 * hb;
            if (row < N) out[(size_t)row * HD + col] = acc[t][i] + bb;
        }
    }
}

// ---------------------------------------------------------------------------
// Edge phase: one edge per wave32. lane covers float4 [4*lane, 4*lane+4) of the
// 128-wide Q/K/V rows; head = lane/4. 4-lane xor-shuffle reduction produces the
// per-head score in every lane of its group, so the V scatter needs no extra
// broadcast. Scatter via global_atomic_add_f32 into out (wV) and z.
// Q/K/V (76.8 MB) are L2-resident (192 MB L2), so gathers hit L2.
// ---------------------------------------------------------------------------
__global__ __launch_bounds__(256) void edge_kernel(
    const float* __restrict__ Qm, const float* __restrict__ Km,
    const float* __restrict__ Vm,
    const int* __restrict__ src, const int* __restrict__ dst,
    float* __restrict__ z, float* __restrict__ out, int E)
{
    const int lane = threadIdx.x & 31;
    const int wave = threadIdx.x >> 5;
    const int head = lane >> 2;

    for (int e = blockIdx.x * 8 + wave; e < E; e += gridDim.x * 8) {
        const int s = src[e];
        const int d = dst[e];

        const float4 kv = ((const float4*)(Km + (size_t)s * HD))[lane];
        const float4 qv = ((const float4*)(Qm + (size_t)d * HD))[lane];

        float p = kv.x * qv.x + kv.y * qv.y + kv.z * qv.z + kv.w * qv.w;
        p += __shfl_xor(p, 1, 32);
        p += __shfl_xor(p, 2, 32);   // all 4 lanes of a head now hold the dot

        float sc = p * 0.25f;                       // / sqrt(D), D = 16
        sc = fminf(5.0f, fmaxf(-5.0f, sc));
        sc = __expf(sc);

        if ((lane & 3) == 0) atomicAdd(&z[(size_t)d * NHEADS + head], sc);

        const float4 vv = ((const float4*)(Vm + (size_t)s * HD))[lane];
        float* ob = out + (size_t)d * HD + lane * 4;
        atomicAdd(ob + 0, vv.x * sc);
        atomicAdd(ob + 1, vv.y * sc);
        atomicAdd(ob + 2, vv.z * sc);
        atomicAdd(ob + 3, vv.w * sc);
    }
}

// ---------------------------------------------------------------------------
// Normalize: out[n,h,d] /= z[n,h]. Flat index i: i/16 == n*8 + h.
// ---------------------------------------------------------------------------
__global__ void norm_kernel(float* __restrict__ out, const float* __restrict__ z,
                            int n) {
    int i = blockIdx.x * blockDim.x + threadIdx.x;
    if (i < n) out[i] /= z[i >> 4];
}

// ---------------------------------------------------------------------------
// Launch
// ---------------------------------------------------------------------------
extern "C" void kernel_launch(void* const* d_in, const int* in_sizes, int n_in,
                              void* d_out, int out_size, void* d_ws, size_t ws_size,
                              hipStream_t stream) {
    const float* h  = (const float*)d_in[0];
    const int*   src = (const int*)d_in[1];
    const int*   dst = (const int*)d_in[2];
    const float* Wq = (const float*)d_in[3];
    const float* bq = (const float*)d_in[4];
    const float* Wk = (const float*)d_in[5];
    const float* bk = (const float*)d_in[6];
    const float* Wv = (const float*)d_in[7];
    const float* bv = (const float*)d_in[8];
    float* out = (float*)d_out;

    const int N = in_sizes[0] / INDIM;
    const int E = in_sizes[1];

    float* Qs = (float*)d_ws;
    float* Ks = Qs + (size_t)N * HD;
    float* Vs = Ks + (size_t)N * HD;
    float* zs = Vs + (size_t)N * HD;
    float* Wt = zs + (size_t)N * NHEADS;   // 3 * 128 * 128 floats = 192 KB

    const int total = N * HD;

    zero_kernel<<<(total + 255) / 256, 256, 0, stream>>>(zs, out, N * NHEADS, total);

    transpose_w_kernel<<<(3 * INDIM * HD + 255) / 256, 256, 0, stream>>>(Wq, Wk, Wv, Wt);

    const int tiles = (N + 15) / 16;
    dim3 ggrid((tiles + 7) / 8, 3);
    qkv_gemm_kernel<<<ggrid, 256, 0, stream>>>(h, Wt, bq, bk, bv, Qs, Ks, Vs, N);

    edge_kernel<<<(E + 7) / 8, 256, 0, stream>>>(Qs, Ks, Vs, src, dst, zs, out, E);

    norm_kernel<<<(total + 255) / 256, 256, 0, stream>>>(out, zs, total);
}